// GraphConv_40501541601587
// MI455X (gfx1250) — compile-verified
//
#include <hip/hip_runtime.h>

typedef __attribute__((ext_vector_type(2))) float v2f;
typedef __attribute__((ext_vector_type(8))) float v8f;

#define N_NODES   50000
#define N_EDGES   800000
#define IN_FEATS  256
#define OUT_FEATS 64
#define M_TILES   (N_NODES / 16)   // 3125, exact

// ---------------------------------------------------------------------------
// Kernel 1: hw = h @ W  (50000x256) @ (256x64) -> (50000x64), full f32 via
// V_WMMA_F32_16X16X4_F32. 256 threads = 8 waves; each wave owns a 16x64 tile
// (4 accumulators). Weight staged through LDS in two 32KB halves using a
// paired-K interleaved layout so every B fragment is ONE ds_load_b64:
//   lds[(k>>1)*128 + n*2 + (k&1)] = W[k, n]
// ---------------------------------------------------------------------------
__global__ __launch_bounds__(256) void gemm_hw_kernel(
    const float* __restrict__ h,
    const float* __restrict__ W,
    float* __restrict__ hw) {
  __shared__ float ws[64 * 128];  // 32 KB: 128 K-rows (64 K-pairs) x 64 cols

  const int tid    = threadIdx.x;
  const int wave   = tid >> 5;
  const int lane   = tid & 31;
  const int laneLo = lane & 15;
  const int kOff   = (lane < 16) ? 0 : 2;   // upper half-wave holds K+2/K+3

  const int tile = blockIdx.x * 8 + wave;   // 16-row output tile
  const bool active = (tile < M_TILES);
  const int rowA = tile * 16 + laneLo;

  v8f acc0 = {}, acc1 = {}, acc2 = {}, acc3 = {};

  for (int kh = 0; kh < 2; ++kh) {
    // Stage W[kh*128 .. kh*128+127, :] into LDS, K-pair interleaved.
    // Each thread: 2 coalesced global b32 loads -> 1 conflict-free ds_store_b64.
    #pragma unroll
    for (int j = 0; j < 16; ++j) {
      const int idx = tid + j * 256;        // 0..4095
      const int kp  = idx >> 6;             // K-pair 0..63
      const int n   = idx & 63;             // column
      const float* g = W + (size_t)(kh * 128 + kp * 2) * OUT_FEATS + n;
      float2 t;
      t.x = g[0];
      t.y = g[OUT_FEATS];
      *(float2*)(ws + kp * 128 + n * 2) = t;
    }
    __syncthreads();

    if (active) {
      const float* hrow = h + (size_t)rowA * IN_FEATS + kh * 128 + kOff;
      for (int k = 0; k < 128; k += 4) {
        // A fragment: lane L<16 -> h[row, k..k+1]; L>=16 -> h[row, k+2..k+3]
        v2f a;
        {
          float2 t = *(const float2*)(hrow + k);
          a.x = t.x; a.y = t.y;
        }
        // B fragments: single ds_load_b64 each (paired-K layout).
        const float* wb = ws + ((k + kOff) >> 1) * 128 + laneLo * 2;
        const v2f b0 = *(const v2f*)(wb + 0);
        const v2f b1 = *(const v2f*)(wb + 32);
        const v2f b2 = *(const v2f*)(wb + 64);
        const v2f b3 = *(const v2f*)(wb + 96);
        acc0 = __builtin_amdgcn_wmma_f32_16x16x4_f32(false, a, false, b0,
                                                     (short)0, acc0, false, false);
        acc1 = __builtin_amdgcn_wmma_f32_16x16x4_f32(false, a, false, b1,
                                                     (short)0, acc1, false, false);
        acc2 = __builtin_amdgcn_wmma_f32_16x16x4_f32(false, a, false, b2,
                                                     (short)0, acc2, false, false);
        acc3 = __builtin_amdgcn_wmma_f32_16x16x4_f32(false, a, false, b3,
                                                     (short)0, acc3, false, false);
      }
    }
    __syncthreads();
  }

  if (active) {
    // C/D layout: VGPR v -> M = v + (lane<16 ? 0 : 8), N = laneLo
    const int mBase = tile * 16 + ((lane < 16) ? 0 : 8);
    float* outp = hw + (size_t)mBase * OUT_FEATS + laneLo;
    #pragma unroll
    for (int v = 0; v < 8; ++v) {
      outp[(size_t)v * OUT_FEATS + 0]  = acc0[v];
      outp[(size_t)v * OUT_FEATS + 16] = acc1[v];
      outp[(size_t)v * OUT_FEATS + 32] = acc2[v];
      outp[(size_t)v * OUT_FEATS + 48] = acc3[v];
    }
  }
}

// ---------------------------------------------------------------------------
// Kernel 2: per-edge scatter: agg[dst] += hw[src] * norm[src].
// 32 threads per edge, 2 columns each (float2 load, 2 f32 atomics).
// hw (12.8MB) and agg (12.8MB) are L2-resident on MI455X (192MB L2), so this
// runs at L2 atomic throughput; HBM traffic is only the edge lists.
// ---------------------------------------------------------------------------
__global__ __launch_bounds__(256) void edge_scatter_kernel(
    const float* __restrict__ hw,
    const float* __restrict__ norm,
    const int* __restrict__ src,
    const int* __restrict__ dst,
    float* __restrict__ agg) {
  const long long gid = (long long)blockIdx.x * blockDim.x + threadIdx.x;
  const int e  = (int)(gid >> 5);
  const int c2 = ((int)gid & 31) * 2;
  if (e >= N_EDGES) return;
  const int s = src[e];
  const int d = dst[e];
  const float ns = norm[s];
  const float2 v = *(const float2*)(hw + (size_t)s * OUT_FEATS + c2);
  float* ap = agg + (size_t)d * OUT_FEATS + c2;
  atomicAdd(ap,     v.x * ns);
  atomicAdd(ap + 1, v.y * ns);
}

// ---------------------------------------------------------------------------
// Kernel 3: out = relu(agg * norm[dstnode] + bias), in place over d_out.
// ---------------------------------------------------------------------------
__global__ __launch_bounds__(256) void finalize_kernel(
    float* __restrict__ out,
    const float* __restrict__ norm,
    const float* __restrict__ bias) {
  const int gid = blockIdx.x * blockDim.x + threadIdx.x;
  if (gid >= N_NODES * OUT_FEATS) return;
  const int node = gid >> 6;
  const int c    = gid & 63;
  const float v = out[gid] * norm[node] + bias[c];
  out[gid] = fmaxf(v, 0.0f);
}

// ---------------------------------------------------------------------------
extern "C" void kernel_launch(void* const* d_in, const int* in_sizes, int n_in,
                              void* d_out, int out_size, void* d_ws, size_t ws_size,
                              hipStream_t stream) {
  const float* h    = (const float*)d_in[0];   // [N_NODES, IN_FEATS]
  const float* norm = (const float*)d_in[1];   // [N_NODES]
  const int*   src  = (const int*)d_in[2];     // [N_EDGES]
  const int*   dst  = (const int*)d_in[3];     // [N_EDGES]
  const float* W    = (const float*)d_in[4];   // [IN_FEATS, OUT_FEATS]
  const float* bias = (const float*)d_in[5];   // [OUT_FEATS]
  float* out = (float*)d_out;                  // [N_NODES, OUT_FEATS]
  float* hw  = (float*)d_ws;                   // 12.8 MB scratch

  // zero the accumulator (d_out) — capture-safe stream op
  hipMemsetAsync(d_out, 0, (size_t)N_NODES * OUT_FEATS * sizeof(float), stream);

  // 1) projection via f32 WMMA
  const int gemm_blocks = (M_TILES + 7) / 8;  // 391
  gemm_hw_kernel<<<gemm_blocks, 256, 0, stream>>>(h, W, hw);

  // 2) edge scatter with fused src-norm scaling
  const long long edge_threads = (long long)N_EDGES * 32;
  const int edge_blocks = (int)((edge_threads + 255) / 256);  // 100000
  edge_scatter_kernel<<<edge_blocks, 256, 0, stream>>>(hw, norm, src, dst, out);

  // 3) dst-norm scale + bias + relu, in place
  const int fin_blocks = (N_NODES * OUT_FEATS + 255) / 256;   // 12500
  finalize_kernel<<<fin_blocks, 256, 0, stream>>>(out, norm, bias);
}